// OneToOneLinear_15144054686486
// MI455X (gfx1250) — compile-verified
//
#include <hip/hip_runtime.h>

typedef __attribute__((ext_vector_type(2))) float v2f;
typedef __attribute__((ext_vector_type(8))) float v8f;

#define FEATURES 1024
#define BATCH    65536

// out[b,f] = in[b,f] * w[f] + bias[f]   (memory-bound: ~537 MB @ 23.3 TB/s ≈ 23 us floor)
//
// WMMA formulation per 16x16 tile (rows m, cols n of a 16-col tile):
//   D = sum_{j=0..3} A_j * B_j + C
//   A_j[m,k] = in[row_m, c + 4j + k]            (16x4 slab of input)
//   B_j[k,n] = (n == 4j+k) ? w[c+n] : 0          (diagonal slice of diag(w))
//   C[m,n]   = bias[c+n]
//
// A-matrix f32 16x4 VGPR layout (documented): lane L -> row M = L%16,
// VGPR v of half h holds K = 2h + v. So a single b64 load per lane at
// column (c + 4j + 2h) lands the slab directly in A layout: no shuffles.
__global__ __launch_bounds__(256) void scale_shift_wmma(
    const float* __restrict__ in, const float* __restrict__ w,
    const float* __restrict__ bias, float* __restrict__ out)
{
  const int lane = threadIdx.x & 31;
  const int warp = threadIdx.x >> 5;
  const int wgid = blockIdx.x * 8 + warp;   // 0..4095 global wave id
  const int n    = lane & 15;               // A/D row-within-tile == D column lane
  const int h    = lane >> 4;               // lane half

  // Each wave owns one 16-wide column tile for its whole lifetime.
  const int col_tile = wgid & 63;           // 1024/16 = 64 tiles
  const int col      = col_tile * 16 + n;

  const float wv = w[col];
  const float bv = bias[col];

  // Diagonal B_j, built once. f32 4x16 B layout: slot (vgpr v, half h) -> K = 2h+v.
  v2f B[4];
#pragma unroll
  for (int j = 0; j < 4; ++j) {
    B[j].x = (n == 4 * j + 2 * h + 0) ? wv : 0.0f;
    B[j].y = (n == 4 * j + 2 * h + 1) ? wv : 0.0f;
  }

  // Bias broadcast in C/D layout (col = lane%16, all 8 vgprs).
  v8f biasC;
#pragma unroll
  for (int v = 0; v < 8; ++v) biasC[v] = bv;

  // 65536 rows / 32-row panels = 2048 panels; 64 wave-groups take 32 each.
  const int p0 = (wgid >> 6) * 32;

  for (int i = 0; i < 32; ++i) {
    const int rbase = (p0 + i) * 32;
    // Direct A-layout load base: row rbase + n, column c + 2h.
    const float* srcA = in + (size_t)(rbase + n) * FEATURES + col_tile * 16 + 2 * h;

    v2f aA[4], aB[4];                       // tile A: rows rbase..+15, tile B: +16..+31
#pragma unroll
    for (int j = 0; j < 4; ++j) {
      aA[j] = *(const v2f*)(srcA + 4 * j);
      aB[j] = *(const v2f*)(srcA + 16 * FEATURES + 4 * j);   // +64KB imm offset
    }

    v8f accA = __builtin_amdgcn_wmma_f32_16x16x4_f32(
        false, aA[0], false, B[0], (short)0, biasC, false, false);
    v8f accB = __builtin_amdgcn_wmma_f32_16x16x4_f32(
        false, aB[0], false, B[0], (short)0, biasC, false, false);
#pragma unroll
    for (int j = 1; j < 4; ++j) {
      accA = __builtin_amdgcn_wmma_f32_16x16x4_f32(
          false, aA[j], false, B[j], (short)0, accA, false, false);
      accB = __builtin_amdgcn_wmma_f32_16x16x4_f32(
          false, aB[j], false, B[j], (short)0, accB, false, false);
    }

    // D layout: lane L, vgpr v -> row v + 8h, col lane%16. Write-once -> NT stores.
    float* dstA = out + (size_t)(rbase + 8 * h) * FEATURES + col;
    float* dstB = dstA + (size_t)16 * FEATURES;
#pragma unroll
    for (int v = 0; v < 8; ++v) {
      __builtin_nontemporal_store(accA[v], dstA + (size_t)v * FEATURES);
      __builtin_nontemporal_store(accB[v], dstB + (size_t)v * FEATURES);
    }
  }
}

extern "C" void kernel_launch(void* const* d_in, const int* in_sizes, int n_in,
                              void* d_out, int out_size, void* d_ws, size_t ws_size,
                              hipStream_t stream) {
  (void)in_sizes; (void)n_in; (void)out_size; (void)d_ws; (void)ws_size;
  const float* inp = (const float*)d_in[0];
  const float* w   = (const float*)d_in[1];
  const float* b   = (const float*)d_in[2];
  float* out       = (float*)d_out;

  // 512 blocks x 256 threads = 4096 waves: covers 65536x1024 exactly.
  scale_shift_wmma<<<dim3(512), dim3(256), 0, stream>>>(inp, w, b, out);
}